// SemiSparseOctree_72868415144187
// MI455X (gfx1250) — compile-verified
//
#include <hip/hip_runtime.h>
#include <stdint.h>

// ---------------------------------------------------------------------------
// SemiSparseOctree SDF interpolation for MI455X (gfx1250).
//
// Bandwidth-bound gather kernel (~190 B moved / ~100 f32 FLOPs per point):
//   * streaming point/index slices staged LDS-side via the Tensor Data Mover
//     (tensor_load_to_lds + s_wait_tensorcnt) -- CDNA5 async-tensor path
//   * per-voxel tables (~12 MB total) stay resident in the 192 MB L2; gathers
//     use default temporal hints so they cache
//   * results written with non-temporal stores to avoid polluting L2
// ---------------------------------------------------------------------------

typedef __attribute__((ext_vector_type(4))) unsigned int v4u;
typedef __attribute__((ext_vector_type(8))) int          v8i;
typedef __attribute__((ext_vector_type(4))) int          v4i;

#define TILE       512            // points per block
#define NTHREADS   256            // 8 waves (wave32)
#define PPT        (TILE / NTHREADS)
#define RESOLUTION 0.05f

#if defined(__AMDGCN__) && __has_builtin(__builtin_amdgcn_tensor_load_to_lds)
#define USE_TDM 1
#else
#define USE_TDM 0
#endif

#if USE_TDM
// 1-D contiguous DMA of `ndw` dwords from global memory into LDS at byte
// offset `lds_byte_off`.  D# layout per CDNA5 ISA ch.8:
//   group0: [1:0]=count=1, [63:32]=lds_addr, [120:64]=global_addr, [127:126]=type=2
//   group1: [17:16]=data_size(2 -> 4B), [79:48]=tensor_dim0, [111:80]=tensor_dim1,
//           [127:112]=tile_dim0, [143:128]=tile_dim1(0), [207:160]=dim0_stride
//   groups 2/3 (+ trailing group on this toolchain): zero — tile dims beyond 0 unused
__device__ __forceinline__ void tdm_load_dwords(uint32_t lds_byte_off,
                                                const void* gptr,
                                                uint32_t ndw /* <= 65535 */) {
  uint64_t ga = (uint64_t)(uintptr_t)gptr;
  v4u g0;
  g0[0] = 1u;                                              // count=1, user mode
  g0[1] = lds_byte_off;                                    // lds_addr
  g0[2] = (uint32_t)ga;                                    // global_addr[31:0]
  g0[3] = (uint32_t)((ga >> 32) & 0x01FFFFFFu) | (2u << 30); // addr[56:32] | type=2

  v8i g1;
  g1[0] = (int)(2u << 16);                                 // data_size = 4 bytes
  g1[1] = (int)((ndw & 0xFFFFu) << 16);                    // tensor_dim0[15:0]
  g1[2] = (int)(((ndw >> 16) & 0xFFFFu) | (1u << 16));     // tensor_dim0[31:16], tensor_dim1=1
  g1[3] = (int)((ndw & 0xFFFFu) << 16);                    // tile_dim0 = ndw
  g1[4] = 0;                                               // tile_dim1=0, tile_dim2=0
  g1[5] = (int)ndw;                                        // tensor_dim0_stride[31:0]
  g1[6] = 0;
  g1[7] = 0;

  v4i gz4 = {0, 0, 0, 0};
  v8i gz8 = {0, 0, 0, 0, 0, 0, 0, 0};
  // clang-23 / therock-10.0 toolchain: 6-arg form
  // (v4u group0, v8i group1, v4i group2, v4i group3, v8i extra, i32 cpol)
  __builtin_amdgcn_tensor_load_to_lds(g0, g1, gz4, gz4, gz8, /*cpol=*/0);
}
#endif

__global__ __launch_bounds__(NTHREADS)
void octree_sdf_kernel(const float* __restrict__ points,         // [P,3]
                       const int*   __restrict__ voxel_indices,  // [P]
                       const float* __restrict__ voxel_centers,  // [V,3]
                       const int*   __restrict__ vertex_indices, // [V,8]
                       const int*   __restrict__ voxel_sizes,    // [V]
                       const float* __restrict__ sdf_priors,     // [V]
                       const float* __restrict__ grad_priors,    // [V,3]
                       float* __restrict__ out_sdf,              // [P]
                       int*   __restrict__ out_idx,              // [P]
                       int P) {
  __shared__ float s_pts[TILE * 3];
  __shared__ int   s_idx[TILE];

  const int base = (int)blockIdx.x * TILE;
  if (base >= P) return;
  const int npts = (P - base < TILE) ? (P - base) : TILE;

#if USE_TDM
  // One wave issues the async tensor DMA for the whole block's slice; it
  // waits on its own TENSORcnt before signalling the workgroup barrier.
  if (threadIdx.x == 0) {
    tdm_load_dwords((uint32_t)(uintptr_t)&s_pts[0],
                    points + (size_t)base * 3, (uint32_t)(npts * 3));
    tdm_load_dwords((uint32_t)(uintptr_t)&s_idx[0],
                    voxel_indices + base, (uint32_t)npts);
    __builtin_amdgcn_s_wait_tensorcnt(0);
  }
#else
  for (int i = (int)threadIdx.x; i < npts * 3; i += NTHREADS)
    s_pts[i] = points[(size_t)base * 3 + i];
  for (int i = (int)threadIdx.x; i < npts; i += NTHREADS)
    s_idx[i] = voxel_indices[base + i];
#endif
  __syncthreads();

#pragma unroll
  for (int it = 0; it < PPT; ++it) {
    const int l = (int)threadIdx.x + it * NTHREADS;
    if (l < npts) {
      const float px = s_pts[l * 3 + 0];
      const float py = s_pts[l * 3 + 1];
      const float pz = s_pts[l * 3 + 2];
      const int   vidx = s_idx[l];

      // Gather tables: L2-resident (~12 MB << 192 MB), default temporal hints.
      const float* vc = voxel_centers + (size_t)vidx * 3;
      const float cx = vc[0], cy = vc[1], cz = vc[2];

      const v4i* vrow = (const v4i*)(vertex_indices + (size_t)vidx * 8); // 32B aligned
      const v4i viA = vrow[0];
      const v4i viB = vrow[1];

      const float sz  = (float)voxel_sizes[vidx] * RESOLUTION;
      const float inv = 1.0f / sz;
      const float h   = 0.5f * sz;

      const float rx = px - cx, ry = py - cy, rz = pz - cz;
      const float qx = rx * inv, qy = ry * inv, qz = rz * inv;
      // trilinear weights: bit=1 -> t = 0.5+q ; bit=0 -> 1-t = 0.5-q
      const float w1x = 0.5f + qx, w0x = 0.5f - qx;
      const float w1y = 0.5f + qy, w0y = 0.5f - qy;
      const float w1z = 0.5f + qz, w0z = 0.5f - qz;
      // delta = p - corner = r - (bit-0.5)*sz : bit=0 -> r+h ; bit=1 -> r-h
      const float dmx = rx + h, dpx = rx - h;
      const float dmy = ry + h, dpy = ry - h;
      const float dmz = rz + h, dpz = rz - h;

      const int vis[8] = {viA[0], viA[1], viA[2], viA[3],
                          viB[0], viB[1], viB[2], viB[3]};

      float acc = 0.0f;
#pragma unroll
      for (int k = 0; k < 8; ++k) {
        const int bx = k & 1, by = (k >> 1) & 1, bz = (k >> 2) & 1;
        const int vk = vis[k];
        const float w = (bx ? w1x : w0x) * (by ? w1y : w0y) * (bz ? w1z : w0z);
        const float  val = sdf_priors[vk];
        const float* g   = grad_priors + (size_t)vk * 3;
        const float gx = g[0], gy = g[1], gzv = g[2];
        const float dx = bx ? dpx : dmx;
        const float dy = by ? dpy : dmy;
        const float dz = bz ? dpz : dmz;
        float t = fmaf(gx, dx, val);
        t = fmaf(gy, dy, t);
        t = fmaf(gzv, dz, t);
        acc = fmaf(w, t, acc);
      }

      // Write-once outputs: non-temporal so they don't evict the gather tables.
      __builtin_nontemporal_store(acc,  &out_sdf[base + l]);
      __builtin_nontemporal_store(vidx, &out_idx[base + l]);
    }
  }
}

extern "C" void kernel_launch(void* const* d_in, const int* in_sizes, int n_in,
                              void* d_out, int out_size, void* d_ws, size_t ws_size,
                              hipStream_t stream) {
  const float* points         = (const float*)d_in[0];
  const int*   voxel_indices  = (const int*)  d_in[1];
  const float* voxel_centers  = (const float*)d_in[2];
  const int*   vertex_indices = (const int*)  d_in[3];
  const int*   voxel_sizes    = (const int*)  d_in[4];
  const float* sdf_priors     = (const float*)d_in[5];
  const float* grad_priors    = (const float*)d_in[6];

  const int P = in_sizes[1];                 // 2,000,000 points
  float* out_sdf = (float*)d_out;            // output 0: sdf [P] f32
  int*   out_idx = (int*)d_out + P;          // output 1: voxel_indices [P] i32

  const int blocks = (P + TILE - 1) / TILE;
  octree_sdf_kernel<<<blocks, NTHREADS, 0, stream>>>(
      points, voxel_indices, voxel_centers, vertex_indices, voxel_sizes,
      sdf_priors, grad_priors, out_sdf, out_idx, P);
}